// NMT_78297253806591
// MI455X (gfx1250) — compile-verified
//
#include <hip/hip_runtime.h>
#include <math.h>

// ---------------------------------------------------------------------------
// NMT forward (bi-LSTM encoder + Luong-attention LSTM decoder) for MI455X.
// All GEMMs run on the CDNA5 matrix pipe via V_WMMA_F32_16X16X4_F32 (fp32,
// bit-faithful to the fp32 reference). Each wave owns a 16x64 macro-tile
// (4 accumulators sharing one A fragment) and prefetches the next K-panel
// (global_prefetch_b8). Output projection (67 GFLOP) is hoisted out of the
// decoder scan into one big WMMA GEMM + row log-softmax.
// ---------------------------------------------------------------------------

static constexpr int SS = 64;    // src len
static constexpr int TT = 64;    // tgt len
static constexpr int NB = 32;    // batch
static constexpr int EE = 512;   // embed dim
static constexpr int HH = 512;   // hidden
static constexpr int HD = 256;   // per-direction encoder hidden
static constexpr int NV = 32000; // tgt vocab

typedef float v2f __attribute__((ext_vector_type(2)));
typedef float v8f __attribute__((ext_vector_type(8)));

__device__ __forceinline__ float sigf(float x) { return 1.0f / (1.0f + expf(-x)); }

// ---------------------------------------------------------------------------
// fp32 WMMA GEMM: D[M,N] = A[M,K] @ W[N,K]^T (+b1[n]) (+b2[n]) (+Csrc[m,n]),
// optional tanh. One wave owns a 16x64 tile (4 x 16x16 accumulators, shared A
// fragment). Requires M%16==0, N%64==0, K%64==0 (true for every call here).
// Layout per CDNA5 ISA (32-bit A 16x4 / B 4x16 / C/D 16x16):
//   lanes 0-15: K = k+{0,1}; lanes 16-31: K = k+{2,3}; row/col = lane&15.
//   D: m = mt*16 + r + 8*(lane>=16), n = ntile*16 + (lane&15).
// ---------------------------------------------------------------------------
__global__ void gemm_wmma_f32(const float* __restrict__ A, const float* __restrict__ W,
                              const float* __restrict__ b1, const float* __restrict__ b2,
                              const float* __restrict__ Csrc, float* __restrict__ D,
                              int M, int N, int K, int doTanh)
{
    const int wave = (int)((blockIdx.x * blockDim.x + threadIdx.x) >> 5);
    const int lane = (int)(threadIdx.x & 31);
    const int mtiles = M >> 4, ntiles = N >> 6;       // 16 x 64 macro-tiles
    if (wave >= mtiles * ntiles) return;              // whole-wave exit: EXEC all-1s
    const int mt = wave / ntiles;
    const int nt = wave % ntiles;
    const int hi = lane >> 4;                          // 0: K+{0,1}  1: K+{2,3}
    const int l  = lane & 15;

    const float* Arow = A + (size_t)(mt * 16 + l) * K + 2 * hi;
    const float* Wr0  = W + (size_t)(nt * 64 + l) * K + 2 * hi;
    const float* Wr1  = Wr0 + (size_t)16 * K;
    const float* Wr2  = Wr0 + (size_t)32 * K;
    const float* Wr3  = Wr0 + (size_t)48 * K;

    v8f acc0 = {0.f,0.f,0.f,0.f,0.f,0.f,0.f,0.f};
    v8f acc1 = acc0, acc2 = acc0, acc3 = acc0;

    for (int k0 = 0; k0 < K; k0 += 64) {
        if (k0 + 64 < K) {                             // prefetch next K-panel
            __builtin_prefetch(Arow + k0 + 64, 0, 0);
            __builtin_prefetch(Wr0  + k0 + 64, 0, 0);
            __builtin_prefetch(Wr1  + k0 + 64, 0, 0);
            __builtin_prefetch(Wr2  + k0 + 64, 0, 0);
            __builtin_prefetch(Wr3  + k0 + 64, 0, 0);
        }
#pragma unroll
        for (int k = k0; k < k0 + 64; k += 4) {
            v2f a  = *(const v2f*)(Arow + k);          // 8B-aligned (K%4==0)
            v2f w0 = *(const v2f*)(Wr0  + k);
            v2f w1 = *(const v2f*)(Wr1  + k);
            v2f w2 = *(const v2f*)(Wr2  + k);
            v2f w3 = *(const v2f*)(Wr3  + k);
            acc0 = __builtin_amdgcn_wmma_f32_16x16x4_f32(false, a, false, w0, (short)0, acc0, false, false);
            acc1 = __builtin_amdgcn_wmma_f32_16x16x4_f32(false, a, false, w1, (short)0, acc1, false, false);
            acc2 = __builtin_amdgcn_wmma_f32_16x16x4_f32(false, a, false, w2, (short)0, acc2, false, false);
            acc3 = __builtin_amdgcn_wmma_f32_16x16x4_f32(false, a, false, w3, (short)0, acc3, false, false);
        }
    }

    const int nbase = nt * 64 + l;
#define GEMM_EPILOGUE(ACC, J)                                                  \
    {                                                                          \
        const int n = nbase + 16 * (J);                                        \
        float bias = 0.f;                                                      \
        if (b1) bias += b1[n];                                                 \
        if (b2) bias += b2[n];                                                 \
        _Pragma("unroll")                                                      \
        for (int r = 0; r < 8; ++r) {                                          \
            const int m = mt * 16 + r + 8 * hi;                                \
            float v = (ACC)[r] + bias;                                         \
            if (Csrc) v += Csrc[(size_t)m * N + n];                            \
            if (doTanh) v = tanhf(v);                                          \
            D[(size_t)m * N + n] = v;                                          \
        }                                                                      \
    }
    GEMM_EPILOGUE(acc0, 0)
    GEMM_EPILOGUE(acc1, 1)
    GEMM_EPILOGUE(acc2, 2)
    GEMM_EPILOGUE(acc3, 3)
#undef GEMM_EPILOGUE
}

// ---------------------------------------------------------------------------
__global__ void zero_f32(float* __restrict__ p, int n)
{
    int i = blockIdx.x * blockDim.x + threadIdx.x;
    if (i < n) p[i] = 0.f;
}

// emb[s*NB+b][e] = src_emb[src[s*NB+b]][e]
__global__ void gather_src_emb(const int* __restrict__ src, const float* __restrict__ tab,
                               float* __restrict__ outp)
{
    int idx = blockIdx.x * blockDim.x + threadIdx.x;  // SS*NB*EE threads
    int e  = idx % EE;
    int sb = idx / EE;
    outp[idx] = tab[(size_t)src[sb] * EE + e];
}

// Fused bidirectional LSTM cell; fwd writes enc_out[t][b][0:HD], bwd enc_out[SS-1-t][b][HD:2HD]
__global__ void enc_cell(const float* __restrict__ Gf, const float* __restrict__ Gb,
                         float* __restrict__ hf, float* __restrict__ cf,
                         float* __restrict__ hb, float* __restrict__ cb,
                         float* __restrict__ encO, int t)
{
    int idx = blockIdx.x * blockDim.x + threadIdx.x;  // 2*NB*HD
    if (idx >= 2 * NB * HD) return;
    int dir = idx / (NB * HD);
    int r   = idx % (NB * HD);
    int b = r / HD, j = r % HD;
    const float* G = dir ? Gb : Gf;
    float* h = dir ? hb : hf;
    float* c = dir ? cb : cf;
    const float* g = G + (size_t)b * 4 * HD;          // gate order i,f,g,o
    float gi = sigf(g[j]);
    float gg_f = sigf(g[HD + j]);
    float gg = tanhf(g[2 * HD + j]);
    float go = sigf(g[3 * HD + j]);
    float cc = gg_f * c[r] + gi * gg;
    c[r] = cc;
    float hhv = go * tanhf(cc);
    h[r] = hhv;
    int s = dir ? (SS - 1 - t) : t;
    encO[((size_t)s * NB + b) * HH + dir * HD + j] = hhv;
}

__global__ void concat_h0c0(const float* __restrict__ hf, const float* __restrict__ cf,
                            const float* __restrict__ hb, const float* __restrict__ cb,
                            float* __restrict__ h0, float* __restrict__ c0)
{
    int idx = blockIdx.x * blockDim.x + threadIdx.x;  // NB*HH
    if (idx >= NB * HH) return;
    int b = idx / HH, k = idx % HH;
    h0[idx] = (k < HD) ? hf[b * HD + k] : hb[b * HD + k - HD];
    c0[idx] = (k < HD) ? cf[b * HD + k] : cb[b * HD + k - HD];
}

// x_t = concat(tgt_emb[tok_t], av_{t-1});  tok_0 = <s> (=2), else tgt[t-1][b]
__global__ void build_x(const int* __restrict__ tgt, const float* __restrict__ temb,
                        const float* __restrict__ av_prev, float* __restrict__ x, int t)
{
    int idx = blockIdx.x * blockDim.x + threadIdx.x;  // NB*(EE+HH)
    if (idx >= NB * (EE + HH)) return;
    int b = idx / (EE + HH), k = idx % (EE + HH);
    float v;
    if (k < EE) {
        int tok = (t == 0) ? 2 : tgt[(t - 1) * NB + b];
        v = temb[(size_t)tok * EE + k];
    } else {
        v = av_prev[b * HH + (k - EE)];
    }
    x[idx] = v;
}

__global__ void lstm_cell_dec(const float* __restrict__ G, float* __restrict__ h,
                              float* __restrict__ c)
{
    int idx = blockIdx.x * blockDim.x + threadIdx.x;  // NB*HH
    if (idx >= NB * HH) return;
    int b = idx / HH, j = idx % HH;
    const float* g = G + (size_t)b * 4 * HH;
    float gi = sigf(g[j]);
    float gf = sigf(g[HH + j]);
    float gg = tanhf(g[2 * HH + j]);
    float go = sigf(g[3 * HH + j]);
    float cc = gf * c[idx] + gi * gg;
    c[idx] = cc;
    h[idx] = go * tanhf(cc);
}

// scores[b][s] = sum_h q[b][h] * encO[s][b][h]   (one wave per (b,s) dot)
__global__ void attn_scores(const float* __restrict__ q, const float* __restrict__ encO,
                            float* __restrict__ sc)
{
    int gid  = blockIdx.x * blockDim.x + threadIdx.x;
    int wave = gid >> 5;
    int lane = gid & 31;
    if (wave >= NB * SS) return;
    int b = wave / SS, s = wave % SS;
    const float* qb = q + (size_t)b * HH;
    const float* e  = encO + ((size_t)s * NB + b) * HH;
    float acc = 0.f;
    for (int h2 = lane; h2 < HH; h2 += 32) acc += qb[h2] * e[h2];
    for (int off = 16; off > 0; off >>= 1) acc += __shfl_down(acc, off, 32);
    if (lane == 0) sc[wave] = acc;
}

// in-place softmax over S=64 per batch row (one block of 64 threads per row)
__global__ void softmax64(float* __restrict__ sc)
{
    __shared__ float red[SS];
    int b = blockIdx.x, s = threadIdx.x;
    float v = sc[b * SS + s];
    red[s] = v; __syncthreads();
    for (int off = SS / 2; off > 0; off >>= 1) {
        if (s < off) red[s] = fmaxf(red[s], red[s + off]);
        __syncthreads();
    }
    float m = red[0]; __syncthreads();
    float e = expf(v - m);
    red[s] = e; __syncthreads();
    for (int off = SS / 2; off > 0; off >>= 1) {
        if (s < off) red[s] += red[s + off];
        __syncthreads();
    }
    sc[b * SS + s] = e / red[0];
}

// hctx[b] = concat(h[b], ctx[b]) with ctx[b][j] = sum_s align[b][s]*encO[s][b][j]
__global__ void build_hctx(const float* __restrict__ h, const float* __restrict__ al,
                           const float* __restrict__ encO, float* __restrict__ hctx)
{
    int idx = blockIdx.x * blockDim.x + threadIdx.x;  // NB*2*HH
    if (idx >= NB * 2 * HH) return;
    int b = idx / (2 * HH), k = idx % (2 * HH);
    float v;
    if (k < HH) {
        v = h[b * HH + k];
    } else {
        int j = k - HH;
        float acc = 0.f;
        for (int s = 0; s < SS; ++s)
            acc += al[b * SS + s] * encO[((size_t)s * NB + b) * HH + j];
        v = acc;
    }
    hctx[idx] = v;
}

// in-place log-softmax over NV per row; one 256-thread block per row
__global__ void log_softmax_rows(float* __restrict__ logits)
{
    __shared__ float red[256];
    int row = blockIdx.x, tid = threadIdx.x;
    float* pr = logits + (size_t)row * NV;
    float m = -1e30f;
    for (int v = tid; v < NV; v += 256) m = fmaxf(m, pr[v]);
    red[tid] = m; __syncthreads();
    for (int off = 128; off > 0; off >>= 1) {
        if (tid < off) red[tid] = fmaxf(red[tid], red[tid + off]);
        __syncthreads();
    }
    m = red[0]; __syncthreads();
    float s = 0.f;
    for (int v = tid; v < NV; v += 256) s += expf(pr[v] - m);
    red[tid] = s; __syncthreads();
    for (int off = 128; off > 0; off >>= 1) {
        if (tid < off) red[tid] += red[tid + off];
        __syncthreads();
    }
    float lse = m + logf(red[0]);
    for (int v = tid; v < NV; v += 256) pr[v] = pr[v] - lse;
}

// ---------------------------------------------------------------------------
static inline void launch_gemm(const float* A, const float* W, const float* b1,
                               const float* b2, const float* Csrc, float* D,
                               int M, int N, int K, int doTanh, hipStream_t stream)
{
    int waves  = (M >> 4) * (N >> 6);            // one wave per 16x64 macro-tile
    int blocks = (waves + 7) / 8;                // 8 waves (256 threads) per block
    gemm_wmma_f32<<<blocks, 256, 0, stream>>>(A, W, b1, b2, Csrc, D, M, N, K, doTanh);
}

extern "C" void kernel_launch(void* const* d_in, const int* in_sizes, int n_in,
                              void* d_out, int out_size, void* d_ws, size_t ws_size,
                              hipStream_t stream)
{
    (void)in_sizes; (void)n_in; (void)out_size; (void)ws_size;

    const int*   src     = (const int*)  d_in[0];
    const int*   tgt     = (const int*)  d_in[1];
    // d_in[2] = is_train (unused; teacher-forcing path)
    const float* src_emb = (const float*)d_in[3];
    const float* Wih_f   = (const float*)d_in[4];
    const float* Whh_f   = (const float*)d_in[5];
    const float* bih_f   = (const float*)d_in[6];
    const float* bhh_f   = (const float*)d_in[7];
    const float* Wih_b   = (const float*)d_in[8];
    const float* Whh_b   = (const float*)d_in[9];
    const float* bih_b   = (const float*)d_in[10];
    const float* bhh_b   = (const float*)d_in[11];
    const float* tgt_emb = (const float*)d_in[12];
    const float* dWih    = (const float*)d_in[13];
    const float* dWhh    = (const float*)d_in[14];
    const float* dbih    = (const float*)d_in[15];
    const float* dbhh    = (const float*)d_in[16];
    const float* Wa      = (const float*)d_in[17];
    const float* Wc      = (const float*)d_in[18];
    const float* bc      = (const float*)d_in[19];
    const float* Wout    = (const float*)d_in[20];
    const float* bout    = (const float*)d_in[21];
    float* out = (float*)d_out;

    // ---- workspace carve-up (floats) ------------------------------------
    float* p = (float*)d_ws;
    float* emb  = p; p += (size_t)SS * NB * EE;        // src embeddings [S,B,E]
    float* GxF  = p; p += (size_t)SS * NB * 4 * HD;    // fwd input gates  [S*B,4Hd]
    float* GxB  = p; p += (size_t)SS * NB * 4 * HD;    // bwd input gates
    float* encO = p; p += (size_t)SS * NB * HH;        // encoder outputs [S,B,H]
    float* GhF  = p; p += (size_t)NB * 4 * HD;         // fwd step gates
    float* GhB  = p; p += (size_t)NB * 4 * HD;         // bwd step gates
    float* hdec = p; p += (size_t)NB * HH;             // decoder h
    float* cdec = p; p += (size_t)NB * HH;             // decoder c
    float* xbuf = p; p += (size_t)NB * (EE + HH);      // decoder input concat
    float* Gdec = p; p += (size_t)NB * 4 * HH;         // decoder gates
    float* qbuf = p; p += (size_t)NB * HH;             // h @ Wa^T
    float* sc   = p; p += (size_t)NB * SS;             // scores / align
    float* hctx = p; p += (size_t)NB * 2 * HH;         // concat(h, ctx)
    float* zreg = p;                                   // contiguous zero region:
    float* hf   = p; p += (size_t)NB * HD;
    float* cf   = p; p += (size_t)NB * HD;
    float* hb   = p; p += (size_t)NB * HD;
    float* cb   = p; p += (size_t)NB * HD;
    float* AV   = p; p += (size_t)(TT + 1) * NB * HH;  // av ring; slot 0 = av_init (zeros)
    const int nzero = 4 * NB * HD + NB * HH;           // hf,cf,hb,cb + AV slot 0

    // ---- init -----------------------------------------------------------
    zero_f32<<<(nzero + 255) / 256, 256, 0, stream>>>(zreg, nzero);

    // ---- encoder --------------------------------------------------------
    gather_src_emb<<<(SS * NB * EE) / 256, 256, 0, stream>>>(src, src_emb, emb);
    // batched input-gate GEMMs over all timesteps (matrix pipe)
    launch_gemm(emb, Wih_f, bih_f, bhh_f, nullptr, GxF, SS * NB, 4 * HD, EE, 0, stream);
    launch_gemm(emb, Wih_b, bih_b, bhh_b, nullptr, GxB, SS * NB, 4 * HD, EE, 0, stream);

    for (int t = 0; t < SS; ++t) {
        launch_gemm(hf, Whh_f, nullptr, nullptr, GxF + (size_t)t * NB * 4 * HD,
                    GhF, NB, 4 * HD, HD, 0, stream);
        launch_gemm(hb, Whh_b, nullptr, nullptr, GxB + (size_t)(SS - 1 - t) * NB * 4 * HD,
                    GhB, NB, 4 * HD, HD, 0, stream);
        enc_cell<<<(2 * NB * HD + 255) / 256, 256, 0, stream>>>(GhF, GhB, hf, cf, hb, cb, encO, t);
    }
    concat_h0c0<<<(NB * HH + 255) / 256, 256, 0, stream>>>(hf, cf, hb, cb, hdec, cdec);

    // ---- decoder (av for each step stored; logp hoisted out of the scan) --
    for (int t = 0; t < TT; ++t) {
        build_x<<<(NB * (EE + HH)) / 256, 256, 0, stream>>>(
            tgt, tgt_emb, AV + (size_t)t * NB * HH, xbuf, t);
        launch_gemm(xbuf, dWih, dbih, dbhh, nullptr, Gdec, NB, 4 * HH, EE + HH, 0, stream);
        launch_gemm(hdec, dWhh, nullptr, nullptr, Gdec, Gdec, NB, 4 * HH, HH, 0, stream);
        lstm_cell_dec<<<(NB * HH) / 256, 256, 0, stream>>>(Gdec, hdec, cdec);
        launch_gemm(hdec, Wa, nullptr, nullptr, nullptr, qbuf, NB, HH, HH, 0, stream);
        attn_scores<<<(NB * SS * 32) / 256, 256, 0, stream>>>(qbuf, encO, sc);
        softmax64<<<NB, SS, 0, stream>>>(sc);
        build_hctx<<<(NB * 2 * HH) / 256, 256, 0, stream>>>(hdec, sc, encO, hctx);
        launch_gemm(hctx, Wc, bc, nullptr, nullptr, AV + (size_t)(t + 1) * NB * HH,
                    NB, HH, 2 * HH, 1, stream);  // fused tanh -> av_t
    }

    // ---- output projection: one big WMMA GEMM (67 GFLOP) + log-softmax ---
    launch_gemm(AV + (size_t)NB * HH, Wout, bout, nullptr, nullptr, out,
                TT * NB, NV, HH, 0, stream);
    log_softmax_rows<<<TT * NB, 256, 0, stream>>>(out);
}